// TurboQuantMSE_30554397343821
// MI455X (gfx1250) — compile-verified
//
#include <hip/hip_runtime.h>
#include <hip/hip_bf16.h>
#include <math.h>

#define BLOCK_LEN 128
#define EPSF 1e-10f

typedef float v2f __attribute__((ext_vector_type(2)));
typedef float v8f __attribute__((ext_vector_type(8)));

// -------- WMMA helper: D(16x16,f32) = A(16x4,f32) x B(4x16,f32) + C --------
static __device__ __forceinline__ v8f wmma4(v2f a, v2f b, v8f c) {
    return __builtin_amdgcn_wmma_f32_16x16x4_f32(
        false, a, false, b, (short)0, c, false, false);
}

// ---------------------------------------------------------------------------
// Prologue 1: fused rotation matrix  M = Hn*D1*Hn*D0
//   M[i][j] = (1/128) * s0[j] * sum_k (-1)^(popc(i&k)+popc(k&j)) * s1[k]
// rotate(x) = M x ; rotate_inv(y) = M^T y
// ---------------------------------------------------------------------------
__global__ void tq_build_M(const float* __restrict__ signs, float* __restrict__ M) {
    int idx = blockIdx.x * blockDim.x + threadIdx.x;   // 0..16383
    int i = idx >> 7, j = idx & 127;
    float acc = 0.f;
    for (int k = 0; k < 128; ++k) {
        int par = (__popc(i & k) + __popc(k & j)) & 1;
        acc += (par ? -1.f : 1.f) * signs[128 + k];
    }
    M[idx] = acc * signs[j] * (1.0f / 128.0f);
}

// ---------------------------------------------------------------------------
// Prologue 2: swizzle M / M^T into per-lane B-fragment layout, **s-major**:
// index = (s*8 + t)*32 + lane, lane holds float2(B[4s+half][n], B[4s+2+half][n])
// with n = 16t + (lane&15).  s-major => the 8 tiles of one K-slice are 8
// consecutive 256B records sharing one base address (immediate offsets).
// ---------------------------------------------------------------------------
__global__ void tq_build_frags(const float* __restrict__ M,
                               v2f* __restrict__ fragMT,   // B = M^T (forward)
                               v2f* __restrict__ fragM) {  // B = M   (inverse)
    int idx  = blockIdx.x * blockDim.x + threadIdx.x;      // 0..8191
    int lane = idx & 31;
    int t    = (idx >> 5) & 7;
    int s    = idx >> 8;
    int k0 = 4 * s + (lane >> 4);
    int n  = 16 * t + (lane & 15);
    v2f mt, mm;
    mt.x = M[n * 128 + k0];        // M^T[k0][n]
    mt.y = M[n * 128 + k0 + 2];    // M^T[k0+2][n]
    mm.x = M[k0 * 128 + n];        // M[k0][n]
    mm.y = M[(k0 + 2) * 128 + n];  // M[k0+2][n]
    fragMT[idx] = mt;
    fragM[idx]  = mm;
}

// ---------------------------------------------------------------------------
// 16x128 row-matrix times 128x128 (fragment form) -> 8 accumulator tiles.
// s outer / t inner + double-buffered B: 8 loads in flight, 8 independent
// accumulation chains (ILP for the matrix pipe), load latency overlapped.
// ---------------------------------------------------------------------------
static __device__ __forceinline__ void mm16x128(const v2f a[32],
                                                const v2f* __restrict__ frag,
                                                int lane, v8f acc[8]) {
    const v2f* bp = frag + lane;
    v2f b[8], bn[8];
#pragma unroll
    for (int t = 0; t < 8; ++t) {
        b[t] = bp[t * 32];
        acc[t] = (v8f){0.f, 0.f, 0.f, 0.f, 0.f, 0.f, 0.f, 0.f};
    }
#pragma unroll
    for (int s = 0; s < 32; ++s) {
        if (s + 1 < 32) {
#pragma unroll
            for (int t = 0; t < 8; ++t)
                bn[t] = bp[((s + 1) * 8 + t) * 32];   // prefetch next slice
        }
#pragma unroll
        for (int t = 0; t < 8; ++t)
            acc[t] = wmma4(a[s], b[t], acc[t]);
#pragma unroll
        for (int t = 0; t < 8; ++t)
            b[t] = bn[t];                             // rename (unrolled)
    }
}

// ---------------------------------------------------------------------------
// Main kernel: 128 threads = 4 waves; each wave handles 16 vectors of len 128.
// Requires nVec % 64 == 0 (true: 2*8*8192 = 131072).
// ---------------------------------------------------------------------------
__global__ __launch_bounds__(128)
void tq_main(const float* __restrict__ x,
             const float* __restrict__ codebook,
             const v2f* __restrict__ fragMT,
             const v2f* __restrict__ fragM,
             float* __restrict__ out) {
    __shared__ float s_bound[256];
    __shared__ float s_code[256];
    __shared__ float s_q[4][16 * 128];   // per-wave D->A transpose buffer

    const int tid = threadIdx.x;
    for (int i = tid; i < 256; i += 128) {
        float c0 = codebook[i];
        s_code[i]  = c0;
        s_bound[i] = (i < 255) ? 0.5f * (c0 + codebook[i + 1]) : 3.0e38f;
    }
    __syncthreads();

    const int lane   = tid & 31;
    const int wid    = tid >> 5;
    const int laneLo = lane & 15;
    const int half   = lane >> 4;

    const long long batch = (long long)blockIdx.x * 4 + wid;   // 16 vectors
    const float* xb = x   + batch * 16 * 128;
    float*       ob = out + batch * 16 * 128;

    // ---- Phase A: load A-fragments of U (16x128), per-vector norms
    v2f a[32];
    float ss = 0.f;
    const float* vp = xb + laneLo * 128 + half * 2;
#pragma unroll
    for (int s = 0; s < 32; ++s) {
        a[s] = *(const v2f*)(vp + 4 * s);
        ss += a[s].x * a[s].x + a[s].y * a[s].y;
    }
    ss += __shfl_xor(ss, 16, 32);
    float nrm = sqrtf(ss);
    float inv = 1.f / (nrm + EPSF);
#pragma unroll
    for (int s = 0; s < 32; ++s) a[s] *= inv;

    // norms redistributed to D-layout (vector m = half*8+g)
    float nv[8];
#pragma unroll
    for (int g = 0; g < 8; ++g) nv[g] = __shfl(nrm, (half << 3) + g, 32);

    // ---- Phase B: R = U * M^T
    v8f acc[8];
    mm16x128(a, fragMT, lane, acc);

    // ---- Phase C: rms per vector (butterfly over 16 lanes in each half)
    float invr[8];
#pragma unroll
    for (int g = 0; g < 8; ++g) {
        float v = 0.f;
#pragma unroll
        for (int t = 0; t < 8; ++t) { float e = acc[t][g]; v += e * e; }
        v += __shfl_xor(v, 1, 32);
        v += __shfl_xor(v, 2, 32);
        v += __shfl_xor(v, 4, 32);
        v += __shfl_xor(v, 8, 32);
        invr[g] = 1.f / (sqrtf(v * (1.0f / 128.0f)) + EPSF);
    }

    // ---- Phase D: quantize + <r,q>, <q,q>; scatter q to LDS [m][n]
    float num[8], den[8];
#pragma unroll
    for (int g = 0; g < 8; ++g) { num[g] = 0.f; den[g] = 0.f; }
    float* sq = &s_q[wid][0];
#pragma unroll
    for (int t = 0; t < 8; ++t) {
#pragma unroll
        for (int g = 0; g < 8; ++g) {
            float r  = acc[t][g];
            float xn = r * invr[g];
            int lo = 0, hi = 255;
#pragma unroll
            for (int it = 0; it < 8; ++it) {
                int mid = (lo + hi) >> 1;
                bool cnd = s_bound[mid] < xn;   // searchsorted 'left'
                lo = cnd ? mid + 1 : lo;
                hi = cnd ? hi : mid;
            }
            float qv = s_code[lo];
            num[g] += r * qv;
            den[g] += qv * qv;
            sq[(half * 8 + g) * 128 + t * 16 + laneLo] = qv;
        }
    }
    float scl[8];
#pragma unroll
    for (int g = 0; g < 8; ++g) {
        float n2 = num[g], d2 = den[g];
        n2 += __shfl_xor(n2, 1, 32);  d2 += __shfl_xor(d2, 1, 32);
        n2 += __shfl_xor(n2, 2, 32);  d2 += __shfl_xor(d2, 2, 32);
        n2 += __shfl_xor(n2, 4, 32);  d2 += __shfl_xor(d2, 4, 32);
        n2 += __shfl_xor(n2, 8, 32);  d2 += __shfl_xor(d2, 8, 32);
        scl[g] = (n2 / (d2 + EPSF)) * nv[g];    // gamma * ||x||
    }

    __syncthreads();   // order LDS scatter before transposed reload

    // ---- Phase E: reload Q as A-fragments
    v2f qa[32];
    const float* qp = sq + laneLo * 128 + half * 2;
#pragma unroll
    for (int s = 0; s < 32; ++s)
        qa[s] = *(const v2f*)(qp + 4 * s);

    // ---- Phase F: Z = Q * M
    v8f zac[8];
    mm16x128(qa, fragM, lane, zac);

    // ---- Phase G: scale & store
#pragma unroll
    for (int t = 0; t < 8; ++t)
#pragma unroll
        for (int g = 0; g < 8; ++g)
            ob[(half * 8 + g) * 128 + t * 16 + laneLo] = zac[t][g] * scl[g];
}

// ---------------------------------------------------------------------------
extern "C" void kernel_launch(void* const* d_in, const int* in_sizes, int n_in,
                              void* d_out, int out_size, void* d_ws, size_t ws_size,
                              hipStream_t stream) {
    const float* x        = (const float*)d_in[0];   // (2,8,8192,128) f32
    const float* signs    = (const float*)d_in[1];   // (2,128) f32
    const float* codebook = (const float*)d_in[2];   // (256,) f32
    float* out = (float*)d_out;

    // workspace: M (64KB) | fragMT (64KB) | fragM (64KB)
    float* wsf    = (float*)d_ws;
    float* M      = wsf;
    v2f*   fragMT = (v2f*)(wsf + 16384);
    v2f*   fragM  = (v2f*)(wsf + 32768);

    int nVec = in_sizes[0] / BLOCK_LEN;              // 131072

    tq_build_M    <<<64, 256, 0, stream>>>(signs, M);
    tq_build_frags<<<32, 256, 0, stream>>>(M, fragMT, fragM);

    int nBlocks = nVec / 64;                         // 4 waves x 16 vectors
    tq_main<<<nBlocks, 128, 0, stream>>>(x, codebook, fragMT, fragM, out);
}